// TripletLossNew_9337258902044
// MI455X (gfx1250) — compile-verified
//
#include <hip/hip_runtime.h>
#include <hip/hip_bf16.h>
#include <cfloat>

typedef float v2f __attribute__((ext_vector_type(2)));
typedef float v4f __attribute__((ext_vector_type(4)));
typedef float v8f __attribute__((ext_vector_type(8)));

#define TL_MARGIN 0.3f
#define TL_EPS    1e-6f
#define NK        8      // K-dimension split (waves per row-group)

// ---------------------------------------------------------------------------
// Kernel 1: per-row sum of (x1 - x2 + eps)^2, reduction done on the matrix
// pipe via V_WMMA_F32_16X16X4_F32 with an all-ones B operand.
//
// Wave mapping (wave32): lane L -> data row m = L%16 of its 16-row group,
// half h = L/16. A-fragment layout (ISA 7.12.2, 32-bit A 16x4):
//   VGPR v holds A[m][2h+v].
// Each lane loads a contiguous float4 (cols c0+4h .. c0+4h+3). WMMA #1
// consumes elements {0,1} (mapped to K=2h,2h+1), WMMA #2 consumes {2,3}.
// Across both halves + both WMMAs, 8 distinct columns are summed exactly
// once — K-slot assignment is a permutation, and sums are order-invariant.
// With B[k][n] == 1.0 for all k,n (layout-invariant), every column of the
// f32 16x16 accumulator carries the running row sum:
//   C/D layout: VGPR j, lanes 0-15 -> M=j, lanes 16-31 -> M=j+8.
// ---------------------------------------------------------------------------
__global__ void __launch_bounds__(256)
rowsq_wmma_kernel(const float* __restrict__ in1,
                  const float* __restrict__ in2,
                  float* __restrict__ partial,   // [NK][nrows]
                  int nrows, int d) {
    const int tid  = blockIdx.x * 256 + threadIdx.x;
    const int wid  = tid >> 5;            // global wave id
    const int lane = threadIdx.x & 31;

    const int kc = wid & (NK - 1);        // K-chunk index
    const int rg = wid / NK;              // 16-row group index
    const int m  = lane & 15;
    const int h  = lane >> 4;

    const int  kchunk = d / NK;           // 256 columns per chunk
    const long long row = (long long)rg * 16 + m;

    const float* p1 = in1 + row * (long long)d + (long long)kc * kchunk + 4 * h;
    const float* p2 = in2 + row * (long long)d + (long long)kc * kchunk + 4 * h;

    v8f acc = {};                          // f32 16x16 accumulator (8 VGPRs)
    v2f bones = {1.0f, 1.0f};              // B = all-ones (layout-invariant)

    const int iters = kchunk >> 3;         // 8 columns consumed per iteration
    for (int it = 0; it < iters; ++it) {
        v4f a = *(const v4f*)p1;
        v4f b = *(const v4f*)p2;
        p1 += 8;                           // 8 data columns / iter (4 per half)
        p2 += 8;

        v4f dd;
        dd[0] = (a[0] - b[0]) + TL_EPS;
        dd[1] = (a[1] - b[1]) + TL_EPS;
        dd[2] = (a[2] - b[2]) + TL_EPS;
        dd[3] = (a[3] - b[3]) + TL_EPS;
        v4f sq = dd * dd;

        v2f alo = {sq[0], sq[1]};          // K = 2h, 2h+1
        v2f ahi = {sq[2], sq[3]};          // K = 2h, 2h+1 of the next WMMA

        acc = __builtin_amdgcn_wmma_f32_16x16x4_f32(
                  false, alo, false, bones, (short)0, acc, false, false);
        acc = __builtin_amdgcn_wmma_f32_16x16x4_f32(
                  false, ahi, false, bones, (short)0, acc, false, false);
    }

    // Extract: VGPR j / lane 0  -> rowsum(group row j)
    //          VGPR j / lane 16 -> rowsum(group row j+8)
    float* dst = partial + (size_t)kc * nrows + (size_t)rg * 16;
    if (lane == 0) {
#pragma unroll
        for (int j = 0; j < 8; ++j) dst[j] = acc[j];
    } else if (lane == 16) {
#pragma unroll
        for (int j = 0; j < 8; ++j) dst[8 + j] = acc[j];
    }
}

// ---------------------------------------------------------------------------
// Kernel 2: combine K-split partials, take sqrt -> dist[nrows]
// ---------------------------------------------------------------------------
__global__ void __launch_bounds__(256)
finish_dist_kernel(const float* __restrict__ partial,
                   float* __restrict__ dist, int nrows) {
    int i = blockIdx.x * 256 + threadIdx.x;
    if (i < nrows) {
        float s = 0.0f;
#pragma unroll
        for (int k = 0; k < NK; ++k) s += partial[(size_t)k * nrows + i];
        dist[i] = sqrtf(s);
    }
}

// ---------------------------------------------------------------------------
// Kernel 3: hardest-positive / hardest-negative per row + margin loss mean.
// One block of n (=128) threads; thread i owns row i of the n x n dist matrix.
// ---------------------------------------------------------------------------
__global__ void triplet_loss_kernel(const float* __restrict__ dist,
                                    const int* __restrict__ targets,
                                    float* __restrict__ out, int n) {
    __shared__ int   tsh[256];
    __shared__ float red[256];
    const int i = threadIdx.x;

    if (i < n) tsh[i] = targets[i];
    __syncthreads();

    float loss = 0.0f;
    if (i < n) {
        const int ti = tsh[i];
        float ap = -FLT_MAX;   // diagonal is always in-mask, so this is replaced
        float an =  FLT_MAX;   // stays FLT_MAX if no negatives (matches 'big')
        for (int j = 0; j < n; ++j) {
            float dv = dist[(size_t)i * n + j];
            if (tsh[j] == ti) ap = fmaxf(ap, dv);
            else              an = fminf(an, dv);
        }
        loss = fmaxf(ap - an + TL_MARGIN, 0.0f);
    }
    red[i] = loss;
    __syncthreads();

    if (i == 0) {
        float s = 0.0f;
        for (int j = 0; j < n; ++j) s += red[j];
        out[0] = s / (float)n;
    }
}

// ---------------------------------------------------------------------------
// Launch
// ---------------------------------------------------------------------------
extern "C" void kernel_launch(void* const* d_in, const int* in_sizes, int n_in,
                              void* d_out, int out_size, void* d_ws, size_t ws_size,
                              hipStream_t stream) {
    const float* in1     = (const float*)d_in[0];
    const float* in2     = (const float*)d_in[1];
    const int*   targets = (const int*)d_in[2];

    const int n     = in_sizes[2];              // 128
    const int nrows = n * n;                    // 16384
    const int d     = in_sizes[0] / nrows;      // 2048

    float* partial = (float*)d_ws;              // NK * nrows floats
    float* dist    = partial + (size_t)NK * nrows;

    const int waves  = (nrows / 16) * NK;       // 8192 waves
    const int blocks = (waves * 32) / 256;      // 1024 blocks of 8 waves

    rowsq_wmma_kernel<<<blocks, 256, 0, stream>>>(in1, in2, partial, nrows, d);
    finish_dist_kernel<<<(nrows + 255) / 256, 256, 0, stream>>>(partial, dist, nrows);
    triplet_loss_kernel<<<1, n, 0, stream>>>(dist, targets, (float*)d_out, n);
}